// MyCEALNetwork_23433341567746
// MI455X (gfx1250) — compile-verified
//
#include <hip/hip_runtime.h>
#include <math.h>

#define NN 50000
#define NE 800000
#define FD 64
#define EDIMS 16
#define NG 64
#define NL 2
#define KEDGE 144              // 64 (x[dst]) + 64 (x[src]) + 16 (edge_attr, We folded in)
#define KPOST 832              // 64 (x) + 12*64 (scaled aggregators)
#define BN_EPS 1e-5f
#define STD_EPS 1e-5f

typedef __attribute__((ext_vector_type(2))) float v2f;
typedef __attribute__((ext_vector_type(8))) float v8f;

// D = A(16x4) * B(4x16) + C, f32 everywhere. 8-arg VOP3P form.
static __device__ __forceinline__ v8f wmma4(v2f a, v2f b, v8f c) {
  return __builtin_amdgcn_wmma_f32_16x16x4_f32(false, a, false, b, (short)0, c, false, false);
}

// float atomic min/max via integer atomics (bit-order trick; init min=+INF, max=-INF)
static __device__ __forceinline__ void atomicMaxF(float* addr, float val) {
  if (val >= 0.f) atomicMax((int*)addr, __float_as_int(val));
  else            atomicMin((unsigned int*)addr, __float_as_uint(val));
}
static __device__ __forceinline__ void atomicMinF(float* addr, float val) {
  if (val >= 0.f) atomicMin((int*)addr, __float_as_int(val));
  else            atomicMax((unsigned int*)addr, __float_as_uint(val));
}

__global__ void k_fill(float* p, float v, size_t n) {
  size_t i = (size_t)blockIdx.x * blockDim.x + threadIdx.x;
  size_t stride = (size_t)gridDim.x * blockDim.x;
  for (; i < n; i += stride) p[i] = v;
}

__global__ void k_deg(const int* __restrict__ dst, float* __restrict__ deg) {
  int i = blockIdx.x * blockDim.x + threadIdx.x;
  int stride = gridDim.x * blockDim.x;
  for (; i < NE; i += stride) atomicAdd(&deg[dst[i]], 1.f);
}

__global__ void k_logsum(const float* __restrict__ deg, float* __restrict__ accum) {
  __shared__ float s[256];
  float v = 0.f;
  for (int i = blockIdx.x * blockDim.x + threadIdx.x; i < NN; i += gridDim.x * blockDim.x)
    v += logf(deg[i] + 1.f);
  s[threadIdx.x] = v;
  __syncthreads();
  for (int st = 128; st > 0; st >>= 1) {
    if (threadIdx.x < st) s[threadIdx.x] += s[threadIdx.x + st];
    __syncthreads();
  }
  if (threadIdx.x == 0) atomicAdd(accum, s[0]);
}

__global__ void k_ampatt(const float* __restrict__ deg, const float* __restrict__ accum,
                         float* __restrict__ amp, float* __restrict__ att) {
  int i = blockIdx.x * blockDim.x + threadIdx.x;
  if (i >= NN) return;
  float avg = accum[0] / (float)NN;
  float logd = logf(fmaxf(deg[i], 1.f) + 1.f);
  amp[i] = logd / avg;
  att[i] = avg / logd;
}

// Fold We into Wpre rows [128:192): Wcat[144][64], bcat[64]
__global__ void k_fold_edge(const float* __restrict__ We, const float* __restrict__ be,
                            const float* __restrict__ Wpre, const float* __restrict__ bpre,
                            float* __restrict__ Wcat, float* __restrict__ bcat) {
  int idx = blockIdx.x * blockDim.x + threadIdx.x;
  if (idx < KEDGE * FD) {
    int k = idx / FD, n = idx % FD;
    float v;
    if (k < 128) {
      v = Wpre[k * FD + n];
    } else {
      int i = k - 128;
      float a = 0.f;
      for (int j = 0; j < FD; j++) a += We[i * FD + j] * Wpre[(128 + j) * FD + n];
      v = a;
    }
    Wcat[idx] = v;
  }
  if (idx < FD) {
    float a = bpre[idx];
    for (int j = 0; j < FD; j++) a += be[j] * Wpre[(128 + j) * FD + idx];
    bcat[idx] = a;
  }
}

// Fold Wpost @ Wlin: W2cat[832][64], b2cat[64]
__global__ void k_fold_post(const float* __restrict__ Wpost, const float* __restrict__ bpost,
                            const float* __restrict__ Wlin, const float* __restrict__ blin,
                            float* __restrict__ W2cat, float* __restrict__ b2cat) {
  int idx = blockIdx.x * blockDim.x + threadIdx.x;
  if (idx < KPOST * FD) {
    int k = idx / FD, n = idx % FD;
    float a = 0.f;
    for (int m = 0; m < FD; m++) a += Wpost[k * FD + m] * Wlin[m * FD + n];
    W2cat[idx] = a;
  }
  if (idx < FD) {
    float a = blin[idx];
    for (int m = 0; m < FD; m++) a += bpost[m] * Wlin[m * FD + idx];
    b2cat[idx] = a;
  }
}

// Fused edge MLP (WMMA, K=144) + scatter of sum/sumsq/min/max into node aggregates.
__global__ void __launch_bounds__(128) k_edge(
    const float* __restrict__ x, const float* __restrict__ ea,
    const int* __restrict__ srcIdx, const int* __restrict__ dstIdx,
    const float* __restrict__ Wcat, const float* __restrict__ bcat,
    float* __restrict__ aggSum, float* __restrict__ aggSq,
    float* __restrict__ aggMin, float* __restrict__ aggMax,
    int tilesPerBlock) {
  __shared__ float sW[KEDGE][FD];        // 36 KB  (weights, resident for all tiles)
  __shared__ float sA[16][KEDGE + 2];    // 9.3 KB (gathered A tile, padded stride)
  __shared__ int sSrc[16];
  __shared__ int sDst[16];

  const int tid = threadIdx.x;
  const int lane = tid & 31;
  const int wave = tid >> 5;           // 4 waves -> 4 x 16 output channels
  const int half = lane >> 4;          // 0: lanes 0-15, 1: lanes 16-31
  const int lr = lane & 15;
  const int koff = half * 2;
  const int nbase = wave * 16;

  for (int i = tid; i < KEDGE * FD; i += 128) sW[i / FD][i % FD] = Wcat[i];
  __syncthreads();

  for (int tt = 0; tt < tilesPerBlock; ++tt) {
    int tile = blockIdx.x * tilesPerBlock + tt;
    int e0 = tile * 16;
    if (e0 >= NE) break;

    if (tid < 16) {
      int e = e0 + tid;
      sSrc[tid] = (e < NE) ? srcIdx[e] : 0;
      sDst[tid] = (e < NE) ? dstIdx[e] : -1;
    }
    __syncthreads();

    for (int i = tid; i < 16 * KEDGE; i += 128) {
      int r = i / KEDGE, c = i % KEDGE;
      int e = e0 + r;
      float v = 0.f;
      if (e < NE) {
        if (c < FD)            v = x[(size_t)sDst[r] * FD + c];
        else if (c < 2 * FD)   v = x[(size_t)sSrc[r] * FD + (c - FD)];
        else                   v = ea[(size_t)e * EDIMS + (c - 2 * FD)];
      }
      sA[r][c] = v;
    }
    __syncthreads();

    v8f acc = {};
    for (int k0 = 0; k0 < KEDGE; k0 += 4) {
      v2f a, b;
      a.x = sA[lr][k0 + koff];
      a.y = sA[lr][k0 + koff + 1];
      b.x = sW[k0 + koff][nbase + lr];
      b.y = sW[k0 + koff + 1][nbase + lr];
      acc = wmma4(a, b, acc);
    }

    float bias = bcat[nbase + lr];
#pragma unroll
    for (int r = 0; r < 8; ++r) {
      int m = half ? (r + 8) : r;
      int e = e0 + m;
      int d = sDst[m];
      if (e < NE && d >= 0) {
        float h = acc[r] + bias;
        size_t off = (size_t)d * FD + nbase + lr;
        atomicAdd(&aggSum[off], h);
        atomicAdd(&aggSq[off], h * h);
        atomicMinF(&aggMin[off], h);
        atomicMaxF(&aggMax[off], h);
      }
    }
    __syncthreads();
  }
}

// Node update: build [x | aggs | amp*aggs | att*aggs] rows in LDS, WMMA against folded Wpost@Wlin.
__global__ void __launch_bounds__(128) k_node(
    const float* __restrict__ x,
    const float* __restrict__ aggSum, const float* __restrict__ aggSq,
    const float* __restrict__ aggMin, const float* __restrict__ aggMax,
    const float* __restrict__ deg, const float* __restrict__ amp, const float* __restrict__ att,
    const float* __restrict__ W2cat, const float* __restrict__ b2cat,
    float* __restrict__ y, int tilesPerBlock) {
  __shared__ float sA[16][KPOST + 2];   // 53.4 KB

  const int tid = threadIdx.x;
  const int lane = tid & 31;
  const int wave = tid >> 5;
  const int half = lane >> 4;
  const int lr = lane & 15;
  const int koff = half * 2;
  const int nbase = wave * 16;

  for (int tt = 0; tt < tilesPerBlock; ++tt) {
    int tile = blockIdx.x * tilesPerBlock + tt;
    int n0 = tile * 16;
    if (n0 >= NN) break;

    for (int i = tid; i < 16 * FD; i += 128) {
      int r = i >> 6, c = i & 63;
      int node = n0 + r;
      float xv = 0.f, mean = 0.f, mn = 0.f, mx = 0.f, sd = 0.f, ampv = 0.f, attv = 0.f;
      if (node < NN) {
        size_t off = (size_t)node * FD + c;
        float dg = deg[node];
        float degc = fmaxf(dg, 1.f);
        mean = aggSum[off] / degc;
        float sq = aggSq[off] / degc;
        sd = sqrtf(fmaxf(sq - mean * mean, 0.f) + STD_EPS);
        mn = (dg > 0.f) ? aggMin[off] : 0.f;
        mx = (dg > 0.f) ? aggMax[off] : 0.f;
        xv = x[off];
        ampv = amp[node];
        attv = att[node];
      }
      sA[r][c]            = xv;
      sA[r][64 + c]       = mean;
      sA[r][128 + c]      = mn;
      sA[r][192 + c]      = mx;
      sA[r][256 + c]      = sd;
      sA[r][320 + c]      = mean * ampv;
      sA[r][384 + c]      = mn * ampv;
      sA[r][448 + c]      = mx * ampv;
      sA[r][512 + c]      = sd * ampv;
      sA[r][576 + c]      = mean * attv;
      sA[r][640 + c]      = mn * attv;
      sA[r][704 + c]      = mx * attv;
      sA[r][768 + c]      = sd * attv;
    }
    __syncthreads();

    v8f acc = {};
    for (int k0 = 0; k0 < KPOST; k0 += 4) {
      v2f a, b;
      a.x = sA[lr][k0 + koff];
      a.y = sA[lr][k0 + koff + 1];
      b.x = W2cat[(size_t)(k0 + koff) * FD + nbase + lr];      // L2-resident weights
      b.y = W2cat[(size_t)(k0 + koff + 1) * FD + nbase + lr];
      acc = wmma4(a, b, acc);
    }

    float bias = b2cat[nbase + lr];
#pragma unroll
    for (int r = 0; r < 8; ++r) {
      int m = half ? (r + 8) : r;
      int node = n0 + m;
      if (node < NN) y[(size_t)node * FD + nbase + lr] = acc[r] + bias;
    }
    __syncthreads();
  }
}

__global__ void k_bnstats(const float* __restrict__ y, float* __restrict__ bnSum,
                          float* __restrict__ bnSq) {
  __shared__ float ssum[FD];
  __shared__ float ssq[FD];
  if (threadIdx.x < FD) { ssum[threadIdx.x] = 0.f; ssq[threadIdx.x] = 0.f; }
  __syncthreads();
  size_t i = (size_t)blockIdx.x * blockDim.x + threadIdx.x;
  size_t stride = (size_t)gridDim.x * blockDim.x;
  for (; i < (size_t)NN * FD; i += stride) {
    float v = y[i];
    int c = (int)(i & 63);
    atomicAdd(&ssum[c], v);
    atomicAdd(&ssq[c], v * v);
  }
  __syncthreads();
  if (threadIdx.x < FD) {
    atomicAdd(&bnSum[threadIdx.x], ssum[threadIdx.x]);
    atomicAdd(&bnSq[threadIdx.x], ssq[threadIdx.x]);
  }
}

__global__ void k_bnapply(const float* __restrict__ y, const float* __restrict__ bnSum,
                          const float* __restrict__ bnSq, const float* __restrict__ gamma,
                          const float* __restrict__ beta, float* __restrict__ xout) {
  size_t i = (size_t)blockIdx.x * blockDim.x + threadIdx.x;
  size_t stride = (size_t)gridDim.x * blockDim.x;
  for (; i < (size_t)NN * FD; i += stride) {
    int c = (int)(i & 63);
    float mu = bnSum[c] / (float)NN;
    float var = bnSq[c] / (float)NN - mu * mu;
    float rs = rsqrtf(var + BN_EPS);
    xout[i] = fmaxf((y[i] - mu) * rs * gamma[c] + beta[c], 0.f);
  }
}

__global__ void k_pool(const float* __restrict__ x, const int* __restrict__ batch,
                       float* __restrict__ gm, float* __restrict__ cnt) {
  size_t i = (size_t)blockIdx.x * blockDim.x + threadIdx.x;
  size_t stride = (size_t)gridDim.x * blockDim.x;
  for (; i < (size_t)NN * FD; i += stride) {
    int node = (int)(i >> 6);
    int c = (int)(i & 63);
    int g = batch[node];
    atomicAdd(&gm[(size_t)g * FD + c], x[i]);
    if (c == 0) atomicAdd(&cnt[g], 1.f);
  }
}

__global__ void k_head(const float* __restrict__ gm, const float* __restrict__ cnt,
                       const float* __restrict__ W1, const float* __restrict__ b1,
                       const float* __restrict__ W2, const float* __restrict__ b2,
                       float* __restrict__ out) {
  int g = threadIdx.x;
  if (g >= NG) return;
  float c = fmaxf(cnt[g], 1.f);
  float m[FD];
  for (int i = 0; i < FD; i++) m[i] = gm[(size_t)g * FD + i] / c;
  float o = b2[0];
  for (int j = 0; j < 40; j++) {
    float h = b1[j];
    for (int i = 0; i < FD; i++) h += m[i] * W1[i * 40 + j];
    o += fmaxf(h, 0.f) * W2[j];
  }
  out[g] = o;
}

extern "C" void kernel_launch(void* const* d_in, const int* in_sizes, int n_in,
                              void* d_out, int out_size, void* d_ws, size_t ws_size,
                              hipStream_t stream) {
  const float* x_in  = (const float*)d_in[0];
  const float* ea    = (const float*)d_in[1];
  const float* We    = (const float*)d_in[2];
  const float* be    = (const float*)d_in[3];
  const float* Wpre  = (const float*)d_in[4];
  const float* bpre  = (const float*)d_in[5];
  const float* Wpost = (const float*)d_in[6];
  const float* bpost = (const float*)d_in[7];
  const float* Wlin  = (const float*)d_in[8];
  const float* blin  = (const float*)d_in[9];
  const float* gamma = (const float*)d_in[10];
  const float* beta  = (const float*)d_in[11];
  const float* W1    = (const float*)d_in[12];
  const float* b1    = (const float*)d_in[13];
  const float* W2    = (const float*)d_in[14];
  const float* b2    = (const float*)d_in[15];
  const int* eidx    = (const int*)d_in[16];
  const int* batch   = (const int*)d_in[17];
  const int* srcIdx = eidx;
  const int* dstIdx = eidx + NE;

  float* p = (float*)d_ws;
  size_t o = 0;
  float* deg   = p + o; o += NN;
  float* amp   = p + o; o += NN;
  float* att   = p + o; o += NN;
  float* scal  = p + o; o += 16;
  float* Wcat  = p + o; o += (size_t)NL * KEDGE * FD;
  float* bcat  = p + o; o += (size_t)NL * FD;
  float* W2cat = p + o; o += (size_t)NL * KPOST * FD;
  float* b2cat = p + o; o += (size_t)NL * FD;
  float* aggSum = p + o; o += (size_t)NN * FD;
  float* aggSq  = p + o; o += (size_t)NN * FD;
  float* aggMin = p + o; o += (size_t)NN * FD;
  float* aggMax = p + o; o += (size_t)NN * FD;
  float* ybuf   = p + o; o += (size_t)NN * FD;
  float* xbuf0  = p + o; o += (size_t)NN * FD;
  float* xbuf1  = p + o; o += (size_t)NN * FD;
  float* bnSum  = p + o; o += FD;
  float* bnSq   = p + o; o += FD;
  float* gm     = p + o; o += (size_t)NG * FD;
  float* cnt    = p + o; o += NG;

  const float INF = __builtin_huge_valf();

  // degree / scalers
  k_fill<<<256, 256, 0, stream>>>(deg, 0.f, (size_t)NN);
  k_fill<<<1, 32, 0, stream>>>(scal, 0.f, (size_t)16);
  k_deg<<<1024, 256, 0, stream>>>(dstIdx, deg);
  k_logsum<<<256, 256, 0, stream>>>(deg, scal);
  k_ampatt<<<(NN + 255) / 256, 256, 0, stream>>>(deg, scal, amp, att);

  // weight folding (per layer)
  for (int l = 0; l < NL; ++l) {
    k_fold_edge<<<(KEDGE * FD + 255) / 256, 256, 0, stream>>>(
        We + (size_t)l * EDIMS * FD, be + (size_t)l * FD,
        Wpre + (size_t)l * 3 * FD * FD, bpre + (size_t)l * FD,
        Wcat + (size_t)l * KEDGE * FD, bcat + (size_t)l * FD);
    k_fold_post<<<(KPOST * FD + 255) / 256, 256, 0, stream>>>(
        Wpost + (size_t)l * KPOST * FD, bpost + (size_t)l * FD,
        Wlin + (size_t)l * FD * FD, blin + (size_t)l * FD,
        W2cat + (size_t)l * KPOST * FD, b2cat + (size_t)l * FD);
  }

  const int edgeTilesPB = 8;
  const int edgeBlocks = (NE / 16 + edgeTilesPB - 1) / edgeTilesPB;   // 6250
  const int nodeTilesPB = 4;
  const int nodeBlocks = (NN / 16 + nodeTilesPB - 1) / nodeTilesPB;   // 782

  for (int l = 0; l < NL; ++l) {
    k_fill<<<2048, 256, 0, stream>>>(aggSum, 0.f, (size_t)NN * FD);
    k_fill<<<2048, 256, 0, stream>>>(aggSq, 0.f, (size_t)NN * FD);
    k_fill<<<2048, 256, 0, stream>>>(aggMin, INF, (size_t)NN * FD);
    k_fill<<<2048, 256, 0, stream>>>(aggMax, -INF, (size_t)NN * FD);
    k_fill<<<1, 128, 0, stream>>>(bnSum, 0.f, (size_t)(2 * FD));  // bnSum+bnSq contiguous

    const float* xl = (l == 0) ? x_in : xbuf0;
    k_edge<<<edgeBlocks, 128, 0, stream>>>(
        xl, ea, srcIdx, dstIdx,
        Wcat + (size_t)l * KEDGE * FD, bcat + (size_t)l * FD,
        aggSum, aggSq, aggMin, aggMax, edgeTilesPB);
    k_node<<<nodeBlocks, 128, 0, stream>>>(
        xl, aggSum, aggSq, aggMin, aggMax, deg, amp, att,
        W2cat + (size_t)l * KPOST * FD, b2cat + (size_t)l * FD, ybuf, nodeTilesPB);
    k_bnstats<<<512, 256, 0, stream>>>(ybuf, bnSum, bnSq);
    float* xo = (l == 0) ? xbuf0 : xbuf1;
    k_bnapply<<<2048, 256, 0, stream>>>(ybuf, bnSum, bnSq,
                                        gamma + (size_t)l * FD, beta + (size_t)l * FD, xo);
  }

  k_fill<<<16, 256, 0, stream>>>(gm, 0.f, (size_t)(NG * FD + NG));  // gm+cnt contiguous
  k_pool<<<2048, 256, 0, stream>>>(xbuf1, batch, gm, cnt);
  k_head<<<1, 64, 0, stream>>>(gm, cnt, W1, b1, W2, b2, (float*)d_out);

  (void)in_sizes; (void)n_in; (void)out_size; (void)ws_size;
}